// QWTForward_57123065037232
// MI455X (gfx1250) — compile-verified
//
#include <hip/hip_runtime.h>
#include <stdint.h>

// ---------------------------------------------------------------------------
// QWT forward, collapsed form:  D = bicubic2x_down(image);  out = 16 scaled
// copies of D (4 LL bands + 12 H bands).  Memory-bound: ~323MB traffic.
//
// Per-wave plan (wave32, one 16x16 D tile per wave):
//   1) async-DMA the 34x34 input halo tile (rows/cols clamped) into LDS
//      via global_load_async_to_lds_b32  (ASYNCcnt)
//   2) horizontal bicubic pass LDS->LDS (34 rows x 16 cols, +2 zero rows)
//   3) vertical pass as 9 chained V_WMMA_F32_16X16X4_F32:
//         D(16x16) = S(16x36) x H(36x16),  S = clamped cubic weights
//   4) D ends in the documented C/D layout (lane=N=x, vgpr=M=y) ->
//      128 coalesced b32 stores per lane cover all 16 output bands.
// ---------------------------------------------------------------------------

typedef float v2f __attribute__((ext_vector_type(2)));
typedef float v8f __attribute__((ext_vector_type(8)));

#define HIN 512
#define WIN 512
#define HO  256
#define WO  256
#define PLANE_IN  (HIN * WIN)
#define PLANE_OUT (HO * WO)
#define NPLANES   64            // B*C = 4*16

#define INP      36             // input tile pitch (34 cols + pad)
#define IN_ELEMS (34 * INP)     // 1224 floats
#define HP       16             // H tile pitch
#define H_ELEMS  (36 * HP)      // 576 floats (rows 34,35 zero pad for K=36)
#define WAVE_LDS (IN_ELEMS + H_ELEMS)   // 1800 floats per wave

__global__ __launch_bounds__(256)
void qwt_kernel(const float* __restrict__ img,
                const float* __restrict__ gl, const float* __restrict__ gh,
                const float* __restrict__ fl, const float* __restrict__ fh,
                float* __restrict__ out)
{
    __shared__ float smem[8 * WAVE_LDS];   // 57.6 KB, private slice per wave

    const int lane = threadIdx.x & 31;
    const int wv   = threadIdx.x >> 5;
    float* IN = &smem[wv * WAVE_LDS];
    float* Hs = IN + IN_ELEMS;

    // filter sums (reference: sum(gl), sum(gh), sum(fl), sum(fh))
    float sg = 0.f, sh = 0.f, sfl = 0.f, sfh = 0.f;
    for (int i = 0; i < 16; ++i) {
        sg  += gl[i];  sh  += gh[i];
        sfl += fl[i];  sfh += fh[i];
    }

    // tile -> (plane, ty, tx)
    const int tile = blockIdx.x * 8 + wv;          // 0 .. 16383
    const int bc   = tile >> 8;                    // plane index 0..63
    const int rem  = tile & 255;
    const int Y0   = (rem >> 4) << 4;              // output tile origin
    const int X0   = (rem & 15) << 4;

    const float* plane = img + (size_t)bc * PLANE_IN;

    // ---- 1) async stage input halo tile: rows clamp(2Y0-1+jr), cols clamp(2X0-1+jc)
    for (int idx = lane; idx < 34 * 34; idx += 32) {
        int jr = idx / 34, jc = idx - jr * 34;
        int gr = 2 * Y0 - 1 + jr;  gr = min(max(gr, 0), HIN - 1);
        int gc = 2 * X0 - 1 + jc;  gc = min(max(gc, 0), WIN - 1);
        unsigned long long ga = (unsigned long long)(const void*)(plane + gr * WIN + gc);
        uint32_t loff = (uint32_t)(uintptr_t)(const void*)(&IN[jr * INP + jc]);
        asm volatile("global_load_async_to_lds_b32 %0, %1, off"
                     :: "v"(loff), "v"(ga) : "memory");
    }
    // zero H pad rows 34,35 (so the 9th WMMA's out-of-range K adds exact 0)
    Hs[(34 + (lane >> 4)) * HP + (lane & 15)] = 0.0f;

    asm volatile("s_wait_asynccnt 0" ::: "memory");

    // ---- 2) horizontal bicubic pass: H[row, xl] over 34 rows x 16 cols
    const float w0 = -0.09375f, w1 = 0.59375f;     // taps [w0 w1 w1 w0]
    for (int i = 0; i < 17; ++i) {
        int idx = i * 32 + lane;                   // 0 .. 543
        int row = idx >> 4, xl = idx & 15;
        const float* rp = &IN[row * INP + 2 * xl];
        Hs[row * HP + xl] = w0 * (rp[0] + rp[3]) + w1 * (rp[1] + rp[2]);
    }

    // ---- 3) vertical pass via WMMA:  D = sum_g A_g(16x4) x B_g(4x16)
    // A layout (ISA 7.12.2, 32-bit 16x4): lane<16 -> K={0,1}, lane>=16 -> K={2,3}; M=lane%16
    // B layout (same K striping, N=lane%16)
    const int l16 = lane & 15;
    const int Klo = (lane < 16) ? 0 : 2;
    v8f acc = {0.f, 0.f, 0.f, 0.f, 0.f, 0.f, 0.f, 0.f};
    for (int g = 0; g < 9; ++g) {
        int t0 = 4 * g + Klo - 2 * l16;            // weight index for K=Klo
        // S[M, j] = w[j - 2M] for 0 <= j-2M <= 3, else 0 (branchless select)
        float a0 = ((t0 == 0) | (t0 == 3)) ? w0 : 0.0f;
        a0       = ((t0 == 1) | (t0 == 2)) ? w1 : a0;
        int t1 = t0 + 1;
        float a1 = ((t1 == 0) | (t1 == 3)) ? w0 : 0.0f;
        a1       = ((t1 == 1) | (t1 == 2)) ? w1 : a1;
        v2f a; a.x = a0; a.y = a1;
        int r0 = 4 * g + Klo;                      // H rows for this K chunk
        v2f b; b.x = Hs[r0 * HP + l16]; b.y = Hs[(r0 + 1) * HP + l16];
        acc = __builtin_amdgcn_wmma_f32_16x16x4_f32(
            /*neg_a=*/false, a, /*neg_b=*/false, b,
            /*c_mod=*/(short)0, acc, /*reuse_a=*/false, /*reuse_b=*/false);
    }

    // ---- 4) stores: C/D layout -> lane holds column x = X0 + l16,
    //         vgpr j holds row y = Y0 + j (+8 for upper half-wave)
    const int bI = bc >> 4, cI = bc & 15;          // batch, channel
    const int x  = X0 + l16;
    const int yb = Y0 + ((lane < 16) ? 0 : 8);

    float* outH = out + (size_t)4 * 64 * PLANE_OUT;   // H region after LL
    for (int i = 0; i < 4; ++i) {
        // band scale factors (see reference LL/b1/b2/b3 tables)
        float aL = (i & 1) ? sfl : sg;
        float aH = (i & 1) ? sfh : sh;
        float bL = (i & 2) ? sfl : sg;
        float bH = (i & 2) ? sfh : sh;
        float sLL = aL * bL;   // LL piece i
        float s0  = aL * bH;   // b1 (m=0)
        float s1  = aH * bL;   // b2 (m=1)
        float s2  = aH * bH;   // b3 (m=2)

        size_t chan = (size_t)(bI * 64 + i * 16 + cI);
        float* pLL = out  + chan * PLANE_OUT;
        float* pH0 = outH + (chan * 3 + 0) * PLANE_OUT;
        float* pH1 = outH + (chan * 3 + 1) * PLANE_OUT;
        float* pH2 = outH + (chan * 3 + 2) * PLANE_OUT;

        #pragma unroll
        for (int j = 0; j < 8; ++j) {
            int off = (yb + j) * WO + x;
            float d = acc[j];
            pLL[off] = d * sLL;
            pH0[off] = d * s0;
            pH1[off] = d * s1;
            pH2[off] = d * s2;
        }
    }
}

extern "C" void kernel_launch(void* const* d_in, const int* in_sizes, int n_in,
                              void* d_out, int out_size, void* d_ws, size_t ws_size,
                              hipStream_t stream)
{
    const float* img = (const float*)d_in[0];
    const float* gl  = (const float*)d_in[1];
    const float* gh  = (const float*)d_in[2];
    const float* fl  = (const float*)d_in[3];
    const float* fh  = (const float*)d_in[4];
    float* out = (float*)d_out;

    // 64 planes x 16x16 tiles of 16x16 outputs = 16384 tiles, 8 tiles/block
    qwt_kernel<<<2048, 256, 0, stream>>>(img, gl, gh, fl, fh, out);
}